// GCNBlock_15109694947953
// MI455X (gfx1250) — compile-verified
//
#include <hip/hip_runtime.h>
#include <hip/hip_bf16.h>
#include <math.h>

typedef float v2f __attribute__((ext_vector_type(2)));
typedef float v8f __attribute__((ext_vector_type(8)));

#define D 128
#define H 16
#define LN_EPS 1e-5f

// ---------------------------------------------------------------------------
// zero fill
// ---------------------------------------------------------------------------
__global__ void k_zero(float* __restrict__ p, long long n) {
    long long i = (long long)blockIdx.x * blockDim.x + threadIdx.x;
    if (i < n) p[i] = 0.0f;
}

// ---------------------------------------------------------------------------
// degree accumulation: deg[dst] += 1 per edge
// ---------------------------------------------------------------------------
__global__ void k_degree(const long long* __restrict__ dst, float* __restrict__ deg,
                         long long E) {
    long long e = (long long)blockIdx.x * blockDim.x + threadIdx.x;
    if (e < E) atomicAdd(&deg[(int)dst[e]], 1.0f);
}

// deg -> dinv = rsqrt(deg + 2)  (in place; deg+2 >= 2 always)
__global__ void k_dinv(float* __restrict__ d, int n) {
    int i = blockIdx.x * blockDim.x + threadIdx.x;
    if (i < n) d[i] = rsqrtf(d[i] + 2.0f);
}

// ---------------------------------------------------------------------------
// GEMM: out[N x 128] = A[N x 128] @ W[128 x 128]  via V_WMMA_F32_16X16X4_F32
// block = 256 threads (8 waves). Block handles 16 rows; wave w handles cols
// [16w, 16w+16). A tile (16x128) staged in LDS; W streamed (L2 resident).
// ---------------------------------------------------------------------------
__global__ void __launch_bounds__(256)
k_gemm128_wmma(const float* __restrict__ A, const float* __restrict__ W,
               float* __restrict__ C) {
    __shared__ float sA[16 * D];  // 8 KB

    const int tid  = threadIdx.x;
    const int wave = tid >> 5;
    const int lane = tid & 31;
    const long long rowBase = (long long)blockIdx.x * 16;

    // cooperative load of the 16x128 A tile (2048 floats = 512 float4)
    {
        const float4* __restrict__ g4 = (const float4*)(A + rowBase * D);
        float4* s4 = (float4*)sA;
        s4[tid]       = g4[tid];
        s4[tid + 256] = g4[tid + 256];
    }
    __syncthreads();

    const int m    = lane & 15;        // row within tile / col within 16-wide slab
    const int hi   = lane >> 4;        // 0: K+0/K+1, 1: K+2/K+3 (ISA lane striping)
    const int col  = (wave << 4) + m;  // output column 0..127
    const int kOff = hi << 1;          // 0 or 2

    v8f acc = {};
#pragma unroll
    for (int k0 = 0; k0 < D; k0 += 4) {
        v2f a, b;
        a.x = sA[m * D + k0 + kOff + 0];
        a.y = sA[m * D + k0 + kOff + 1];
        b.x = W[(k0 + kOff + 0) * D + col];
        b.y = W[(k0 + kOff + 1) * D + col];
        acc = __builtin_amdgcn_wmma_f32_16x16x4_f32(
            /*neg_a=*/false, a, /*neg_b=*/false, b,
            /*c_mod=*/(short)0, acc, /*reuse_a=*/false, /*reuse_b=*/false);
    }

    // C/D layout: VGPR i -> lanes 0..15: M=i, lanes 16..31: M=8+i
    const long long r0 = rowBase + (hi ? 8 : 0);
#pragma unroll
    for (int i = 0; i < 8; ++i) {
        C[(r0 + i) * D + col] = acc[i];
    }
}

// ---------------------------------------------------------------------------
// edge scatter: agg[dst] += xw[src] * (dinv[src]*dinv[dst])
// one lane handles a float4 slice; 32 lanes cover one edge's 128 floats
// ---------------------------------------------------------------------------
__global__ void __launch_bounds__(256)
k_edge_scatter(const float* __restrict__ xw, const float* __restrict__ dinv,
               const long long* __restrict__ src, const long long* __restrict__ dst,
               float* __restrict__ agg, long long E) {
    long long tid = (long long)blockIdx.x * blockDim.x + threadIdx.x;
    long long e = tid >> 5;
    if (e >= E) return;
    const int lane = (int)(tid & 31);
    const int s = (int)src[e];
    const int d = (int)dst[e];
    const float norm = dinv[s] * dinv[d];

    const float4 v = ((const float4*)(xw + (long long)s * D))[lane];
    float* ap = agg + (long long)d * D + lane * 4;
    atomicAdd(ap + 0, v.x * norm);
    atomicAdd(ap + 1, v.y * norm);
    atomicAdd(ap + 2, v.z * norm);
    atomicAdd(ap + 3, v.w * norm);
}

// ---------------------------------------------------------------------------
// per-node: v = agg + xw*(2*dinv^2) + b ; layernorm(g,beta) ; optional relu
// block = 128 threads = one node
// ---------------------------------------------------------------------------
__global__ void __launch_bounds__(128)
k_node_ln(const float* __restrict__ agg, const float* __restrict__ xw,
          const float* __restrict__ dinv, const float* __restrict__ b,
          const float* __restrict__ g, const float* __restrict__ beta,
          float* __restrict__ out, int do_relu) {
    const long long node = blockIdx.x;
    const int d = threadIdx.x;
    __shared__ float red[D];

    float di = dinv[node];
    float c = 2.0f * di * di;
    float v = agg[node * D + d] + xw[node * D + d] * c + b[d];

    // mean
    red[d] = v;
    __syncthreads();
    for (int s = 64; s > 0; s >>= 1) {
        if (d < s) red[d] += red[d + s];
        __syncthreads();
    }
    float mu = red[0] * (1.0f / D);
    __syncthreads();

    // variance
    float dv = v - mu;
    red[d] = dv * dv;
    __syncthreads();
    for (int s = 64; s > 0; s >>= 1) {
        if (d < s) red[d] += red[d + s];
        __syncthreads();
    }
    float var = red[0] * (1.0f / D);

    float y = dv * rsqrtf(var + LN_EPS) * g[d] + beta[d];
    if (do_relu) y = fmaxf(y, 0.0f);
    out[node * D + d] = y;
}

// ---------------------------------------------------------------------------
// SE gate + residual + relu:
//   s = relu(h @ Ws + bs); w = sigmoid(s @ We + be); out = relu(h*w + x)
// block = 128 threads = one node
// ---------------------------------------------------------------------------
__global__ void __launch_bounds__(128)
k_se_out(const float* __restrict__ h2, const float* __restrict__ x,
         const float* __restrict__ Ws, const float* __restrict__ bs,
         const float* __restrict__ We, const float* __restrict__ be,
         float* __restrict__ out) {
    const long long node = blockIdx.x;
    const int d = threadIdx.x;
    __shared__ float sh[D];
    __shared__ float ss[H];

    float h = h2[node * D + d];
    sh[d] = h;
    __syncthreads();

    if (d < H) {
        float acc = bs[d];
#pragma unroll 8
        for (int j = 0; j < D; ++j) acc = fmaf(sh[j], Ws[j * H + d], acc);
        ss[d] = fmaxf(acc, 0.0f);
    }
    __syncthreads();

    float acc = be[d];
#pragma unroll
    for (int j = 0; j < H; ++j) acc = fmaf(ss[j], We[j * D + d], acc);
    float w = 1.0f / (1.0f + __expf(-acc));

    out[node * D + d] = fmaxf(fmaf(h, w, x[node * D + d]), 0.0f);
}

// ---------------------------------------------------------------------------
extern "C" void kernel_launch(void* const* d_in, const int* in_sizes, int n_in,
                              void* d_out, int out_size, void* d_ws, size_t ws_size,
                              hipStream_t stream) {
    const float*     x   = (const float*)d_in[0];
    const long long* ei  = (const long long*)d_in[1];  // [2, E] int64
    const float*     W1  = (const float*)d_in[2];
    const float*     b1  = (const float*)d_in[3];
    const float*     g1  = (const float*)d_in[4];
    const float*     be1 = (const float*)d_in[5];
    const float*     W2  = (const float*)d_in[6];
    const float*     b2  = (const float*)d_in[7];
    const float*     g2  = (const float*)d_in[8];
    const float*     be2 = (const float*)d_in[9];
    const float*     Ws  = (const float*)d_in[10];
    const float*     bs  = (const float*)d_in[11];
    const float*     We  = (const float*)d_in[12];
    const float*     be  = (const float*)d_in[13];
    float* out = (float*)d_out;

    const long long N = in_sizes[0] / D;        // 50000
    const long long E = in_sizes[1] / 2;        // 1.6M
    const long long ND = N * D;

    const long long* src = ei;
    const long long* dst = ei + E;

    float* ws   = (float*)d_ws;
    float* dinv = ws;            // N
    float* xw   = dinv + N;      // N*D
    float* agg  = xw + ND;       // N*D
    float* hbuf = agg + ND;      // N*D

    const int ZB = 256;
    const int rowTiles = (int)(N / 16);                       // 3125
    const int edgeBlocks = (int)((E * 32 + ZB - 1) / ZB);     // 200000

    // degree -> dinv
    k_zero<<<(int)((N + ZB - 1) / ZB), ZB, 0, stream>>>(dinv, N);
    k_degree<<<(int)((E + ZB - 1) / ZB), ZB, 0, stream>>>(dst, dinv, E);
    k_dinv<<<(int)((N + ZB - 1) / ZB), ZB, 0, stream>>>(dinv, (int)N);

    // ---- layer 1 ----
    k_gemm128_wmma<<<rowTiles, 256, 0, stream>>>(x, W1, xw);
    k_zero<<<(int)((ND + ZB - 1) / ZB), ZB, 0, stream>>>(agg, ND);
    k_edge_scatter<<<edgeBlocks, ZB, 0, stream>>>(xw, dinv, src, dst, agg, E);
    k_node_ln<<<(int)N, 128, 0, stream>>>(agg, xw, dinv, b1, g1, be1, hbuf, 1);

    // ---- layer 2 ----
    k_gemm128_wmma<<<rowTiles, 256, 0, stream>>>(hbuf, W2, xw);
    k_zero<<<(int)((ND + ZB - 1) / ZB), ZB, 0, stream>>>(agg, ND);
    k_edge_scatter<<<edgeBlocks, ZB, 0, stream>>>(xw, dinv, src, dst, agg, E);
    k_node_ln<<<(int)N, 128, 0, stream>>>(agg, xw, dinv, b2, g2, be2, hbuf, 0);

    // ---- SE gate + residual ----
    k_se_out<<<(int)N, 128, 0, stream>>>(hbuf, x, Ws, bs, We, be, out);
}